// RegionProposalNetwork_55095840473635
// MI455X (gfx1250) — compile-verified
//
#include <hip/hip_runtime.h>
#include <hip/hip_bf16.h>
#include <stdint.h>

// Problem constants (match reference)
#define BB 8
#define GHH 96
#define GWW 96
#define AA 9
#define NN 32
#define MM (GHH * GWW * AA)   // 82944 anchors per batch
#define APT 4                 // anchors per thread
#define TPB 256               // threads per block
#define APB (APT * TPB)       // 1024 anchors per block; MM % APB == 0 (81 blocks)
#define POS_THRESH 0.7f

typedef uint32_t u32x4 __attribute__((ext_vector_type(4)));
typedef uint32_t u32x8 __attribute__((ext_vector_type(8)));

// ---------------------------------------------------------------------------
// IoU: identical inlined code in both passes -> bit-consistent results for the
// pass-2 equality test against pass-1's per-GT max. Fast reciprocal (1-2 ulp)
// instead of IEEE divide: saves ~8 VALU ops on each of the 21M pairs.
// ---------------------------------------------------------------------------
__device__ __forceinline__ float iou_pair(const float4 a, const float4 g,
                                          float areaA, float areaG) {
    float w = fmaxf(fminf(a.z, g.z) - fmaxf(a.x, g.x), 0.0f);
    float h = fmaxf(fminf(a.w, g.w) - fmaxf(a.y, g.y), 0.0f);
    float inter = w * h;
    return inter * __builtin_amdgcn_rcpf(areaA + areaG - inter);
}

// ---------------------------------------------------------------------------
// Stage the 32 GT boxes (512 B) into LDS with the Tensor Data Mover.
// D# describes a 1-D tile: 128 f32 elements, tensor_dim0 = stride = 128.
// One issue per block (wave 0, EXEC ignored by TDM), TENSORcnt-tracked.
// ---------------------------------------------------------------------------
__device__ __forceinline__ void tdm_stage_gt(float4* sGT, const float4* gbase,
                                             int tid) {
    if (tid == 0) {
        uint32_t lds_addr = (uint32_t)(uintptr_t)sGT;
        uint64_t ga = (uint64_t)(uintptr_t)gbase;
        u32x4 g0;
        g0[0] = 1u;                                    // count=1 (valid user D#)
        g0[1] = lds_addr;                              // lds_addr
        g0[2] = (uint32_t)ga;                          // global_addr[31:0]
        g0[3] = ((uint32_t)(ga >> 32) & 0x01FFFFFFu)   // global_addr[56:32]
                | (2u << 30);                          // type=2 ("image")
        u32x8 g1;
        g1[0] = (2u << 16);                            // data_size=2 -> 4 bytes
        g1[1] = (128u << 16);                          // tensor_dim0 = 128
        g1[2] = (1u << 16);                            // tensor_dim1 = 1
        g1[3] = (128u << 16);                          // tile_dim0 = 128
        g1[4] = 1u;                                    // tile_dim1 = 1
        g1[5] = 128u;                                  // tensor_dim0_stride = 128
        g1[6] = 0u;
        g1[7] = 0u;
        asm volatile("tensor_load_to_lds %0, %1" :: "s"(g0), "s"(g1) : "memory");
        asm volatile("s_wait_tensorcnt 0x0" ::: "memory");
    }
    __syncthreads();
}

// ---------------------------------------------------------------------------
// Stage GT boxes via the ASYNCcnt-tracked global->LDS async-copy path
// (lanes 0..31 of wave 0, wave-uniform branch).
// ---------------------------------------------------------------------------
__device__ __forceinline__ void async_stage_gt(float4* sGT, const float4* gbase,
                                               int tid) {
    if (tid < NN) {
        uint32_t lds_addr = (uint32_t)(uintptr_t)(&sGT[tid]);
        uint32_t voff     = (uint32_t)tid * 16u;
        unsigned long long sbase = (unsigned long long)(uintptr_t)gbase;
        asm volatile("global_load_async_to_lds_b128 %0, %1, %2"
                     :: "v"(lds_addr), "v"(voff), "s"(sbase) : "memory");
        asm volatile("s_wait_asynccnt 0x0" ::: "memory");
    }
    __syncthreads();
}

// ---------------------------------------------------------------------------
// Kernel 0: zero the per-(b,n) global max scratch (B*N = 256 uints in d_ws).
// ---------------------------------------------------------------------------
__global__ void rpn_zero_gmax(unsigned* __restrict__ gmax) {
    gmax[threadIdx.x] = 0u;
}

// ---------------------------------------------------------------------------
// Kernel 1: IoU matrix + per-GT max reduction.
// grid = (MM/APB, B), block = TPB. Each thread owns APT strided anchors.
// Reduction: per-thread max over APT anchors -> shfl_xor wave max ->
// one LDS atomic per wave per n -> one global atomic per (block, n).
// ---------------------------------------------------------------------------
__global__ void rpn_iou_pass1(const float4* __restrict__ anc,
                              const float4* __restrict__ gt,
                              float* __restrict__ iou_out,
                              unsigned* __restrict__ gmax) {
    const int tid = threadIdx.x;
    const int b   = blockIdx.y;
    const int m0  = blockIdx.x * APB + tid;

    __shared__ float4   sGT[NN];
    __shared__ float    sAreaG[NN];
    __shared__ unsigned sMaxBits[NN];   // bit-pattern max valid: IoU >= 0

    if (tid < NN) sMaxBits[tid] = 0u;
    tdm_stage_gt(sGT, gt + (size_t)b * NN, tid);     // includes barrier
    if (tid < NN) {
        float4 g = sGT[tid];
        sAreaG[tid] = (g.z - g.x) * (g.w - g.y);
    }
    __syncthreads();

    float lmax[NN];
#pragma unroll
    for (int n = 0; n < NN; ++n) lmax[n] = 0.0f;

#pragma unroll
    for (int k = 0; k < APT; ++k) {
        const int m = m0 + k * TPB;
        const float4 a = anc[(size_t)b * MM + m];
        const float areaA = (a.z - a.x) * (a.w - a.y);
        float v[NN];
#pragma unroll
        for (int n = 0; n < NN; ++n) {
            float iou = iou_pair(a, sGT[n], areaA, sAreaG[n]);
            v[n] = iou;
            lmax[n] = fmaxf(lmax[n], iou);
        }
        // 128 contiguous bytes per anchor row -> coalesced float4 stores
        float4* out4 = (float4*)(iou_out + ((size_t)b * MM + m) * NN);
#pragma unroll
        for (int q = 0; q < NN / 4; ++q)
            out4[q] = make_float4(v[4 * q], v[4 * q + 1], v[4 * q + 2], v[4 * q + 3]);
    }

    // wave32 max reduction per GT, then one LDS atomic per wave per n
#pragma unroll
    for (int n = 0; n < NN; ++n) {
        float r = lmax[n];
        for (int off = 16; off > 0; off >>= 1)
            r = fmaxf(r, __shfl_xor(r, off, 32));
        lmax[n] = r;
    }
    if ((tid & 31) == 0) {
#pragma unroll
        for (int n = 0; n < NN; ++n)
            atomicMax(&sMaxBits[n], __float_as_uint(lmax[n]));
    }
    __syncthreads();
    if (tid < NN)
        atomicMax(&gmax[b * NN + tid], sMaxBits[tid]);
}

// ---------------------------------------------------------------------------
// Kernel 2: matching + conf / offsets / classes. Recomputes IoU (10.6 MB
// anchor re-read beats an 85 MB IoU matrix re-read).
// ---------------------------------------------------------------------------
__global__ void rpn_match_pass2(const float4* __restrict__ anc,
                                const float4* __restrict__ gt,
                                const int* __restrict__ labels,
                                const unsigned* __restrict__ gmax,
                                float* __restrict__ conf_out,
                                float4* __restrict__ off_out,
                                float* __restrict__ cls_out) {
    const int tid = threadIdx.x;
    const int b   = blockIdx.y;
    const int m0  = blockIdx.x * APB + tid;

    __shared__ float4 sGT[NN];
    __shared__ float  sAreaG[NN];
    __shared__ float  sMax[NN];
    __shared__ int    sLab[NN];

    if (tid < NN) {
        sMax[tid] = __uint_as_float(gmax[b * NN + tid]);
        sLab[tid] = labels[b * NN + tid];
    }
    async_stage_gt(sGT, gt + (size_t)b * NN, tid);   // includes barrier
    if (tid < NN) {
        float4 g = sGT[tid];
        sAreaG[tid] = (g.z - g.x) * (g.w - g.y);
    }
    __syncthreads();

#pragma unroll
    for (int k = 0; k < APT; ++k) {
        const int m = m0 + k * TPB;
        const float4 a = anc[(size_t)b * MM + m];
        const float areaA = (a.z - a.x) * (a.w - a.y);

        float best = -1.0f;  // IoU >= 0 -> n=0 taken; strict '>' keeps FIRST
        int bestn = 0;       // max index (jnp.argmax semantics)
        bool pos = false;
#pragma unroll
        for (int n = 0; n < NN; ++n) {
            float iou = iou_pair(a, sGT[n], areaA, sAreaG[n]);
            if (iou > best) { best = iou; bestn = n; }
            float mg = sMax[n];
            pos = pos || ((iou == mg) && (mg > 0.0f)) || (iou > POS_THRESH);
        }

        const size_t idx = (size_t)b * MM + m;
        const float posf = pos ? 1.0f : 0.0f;

        conf_out[idx] = pos ? best : 0.0f;

        float4 g = sGT[bestn];
        float mcx = (g.x + g.z) * 0.5f, mcy = (g.y + g.w) * 0.5f;
        float mw  = g.z - g.x,          mh  = g.w - g.y;
        float acx = (a.x + a.z) * 0.5f, acy = (a.y + a.w) * 0.5f;
        float aw  = a.z - a.x,          ah  = a.w - a.y;
        float tx = (mcx - acx) / aw;
        float ty = (mcy - acy) / ah;
        float tw = __logf(mw / aw);
        float th = __logf(mh / ah);
        off_out[idx] = make_float4(tx * posf, ty * posf, tw * posf, th * posf);

        cls_out[idx] = pos ? (float)sLab[bestn] : -1.0f;
    }
}

// ---------------------------------------------------------------------------
// Launch: inputs {anchor_boxes f32, bounding_boxes f32, labels i32};
// d_out = [iou | conf | offsets | classes] concatenated flat as f32.
// ---------------------------------------------------------------------------
extern "C" void kernel_launch(void* const* d_in, const int* in_sizes, int n_in,
                              void* d_out, int out_size, void* d_ws, size_t ws_size,
                              hipStream_t stream) {
    (void)in_sizes; (void)n_in; (void)out_size; (void)ws_size;

    const float4* anc    = (const float4*)d_in[0];
    const float4* gt     = (const float4*)d_in[1];
    const int*    labels = (const int*)d_in[2];

    float*  out      = (float*)d_out;
    float*  iou_out  = out;                                   // B*M*N
    float*  conf_out = out + (size_t)BB * MM * NN;            // B*M
    float4* off_out  = (float4*)(conf_out + (size_t)BB * MM); // B*M*4
    float*  cls_out  = (float*)off_out + (size_t)BB * MM * 4; // B*M

    unsigned* gmax = (unsigned*)d_ws;                         // B*N = 256 u32

    rpn_zero_gmax<<<dim3(1), dim3(BB * NN), 0, stream>>>(gmax);

    dim3 grid(MM / APB, BB);
    dim3 block(TPB);
    rpn_iou_pass1<<<grid, block, 0, stream>>>(anc, gt, iou_out, gmax);
    rpn_match_pass2<<<grid, block, 0, stream>>>(anc, gt, labels, gmax,
                                                conf_out, off_out, cls_out);
}